// SDFChamferLoss_22333829939336
// MI455X (gfx1250) — compile-verified
//
#include <hip/hip_runtime.h>
#include <math.h>

#define HH 256
#define WW 256
#define BB 4
#define PMAX 8192
#define EPSF 1e-8f
#define BIGF 1e10f

typedef __attribute__((ext_vector_type(2))) float v2f;
typedef __attribute__((ext_vector_type(4))) float v4f;
typedef __attribute__((ext_vector_type(8))) float v8f;

// ---------------- workspace layout (floats) ----------------
// [0..7]        : int counts per slot (slot = b*2 + {0:pred,1:gt})
// [8..11]       : sum_d1 per sample (pred -> nearest gt)
// [12..15]      : sum_d2 per sample (gt  -> nearest pred)
// [16..19]      : L1 |p-g| sum per sample
// [32 ..]       : pts   8*PMAX*2
// [131104 ..]   : embA  8*PMAX*4   [x, y, x^2+y^2, 1]      (pad 0,0,0,0)
// [393248 ..]   : embB  8*PMAX*4   [-2x, -2y, 1, x^2+y^2]  (pad 0,0,0,BIG)

__global__ void sdfch_init(int* counts, float* accum) {
    int t = threadIdx.x;
    if (t < 8)  counts[t] = 0;
    if (t < 12) accum[t]  = 0.0f;
}

__global__ void sdfch_extract(const float* __restrict__ pred,
                              const float* __restrict__ gt,
                              int* __restrict__ counts,
                              float* __restrict__ pts) {
    const int NV   = (HH - 1) * WW;
    const int NH   = HH * (WW - 1);
    const int CAND = NV + NH;              // 130560 per slot
    int idx  = blockIdx.x * blockDim.x + threadIdx.x;
    int slot = idx / CAND;
    if (slot >= 8) return;
    int c = idx - slot * CAND;
    const float* s = ((slot & 1) ? gt : pred) + (size_t)(slot >> 1) * HH * WW;

    float r, cl;
    bool valid;
    if (c < NV) {                          // vertical pair (i,j)-(i+1,j)
        int i = c / WW, j = c - i * WW;
        float v1 = s[i * WW + j], v2 = s[(i + 1) * WW + j];
        valid = (v1 == 0.0f) | (v2 == 0.0f) | (v1 * v2 < 0.0f);
        float a = fabsf(v1) / (fabsf(v1) + fabsf(v2) + EPSF);
        r  = (v1 == 0.0f) ? (float)i : ((v2 == 0.0f) ? (float)(i + 1) : (float)i + a);
        cl = (float)j;
    } else {                               // horizontal pair (i,j)-(i,j+1)
        int ch = c - NV;
        int i = ch / (WW - 1), j = ch - i * (WW - 1);
        float h1 = s[i * WW + j], h2 = s[i * WW + j + 1];
        valid = (h1 == 0.0f) | (h2 == 0.0f) | (h1 * h2 < 0.0f);
        float a = fabsf(h1) / (fabsf(h1) + fabsf(h2) + EPSF);
        cl = (h1 == 0.0f) ? (float)j : ((h2 == 0.0f) ? (float)(j + 1) : (float)j + a);
        r  = (float)i;
    }
    if (valid) {
        int p = atomicAdd(&counts[slot], 1);
        if (p < PMAX) {
            pts[((size_t)slot * PMAX + p) * 2]     = r;
            pts[((size_t)slot * PMAX + p) * 2 + 1] = cl;
        }
    }
}

// Precompute WMMA-ready operand embeddings, fully padded to PMAX.
__global__ void sdfch_embed(const int* __restrict__ counts,
                            const float* __restrict__ pts,
                            float* __restrict__ embA,
                            float* __restrict__ embB) {
    int idx  = blockIdx.x * 256 + threadIdx.x;     // 8*PMAX threads
    int slot = idx >> 13;
    int n    = idx & (PMAX - 1);
    int c    = min(counts[slot], PMAX);
    v4f ea, eb;
    if (n < c) {
        float x = pts[(size_t)idx * 2];
        float y = pts[(size_t)idx * 2 + 1];
        float s2 = x * x + y * y;
        ea = (v4f){x, y, s2, 1.0f};
        eb = (v4f){-2.0f * x, -2.0f * y, 1.0f, s2};
    } else {
        ea = (v4f){0.0f, 0.0f, 0.0f, 0.0f};
        eb = (v4f){0.0f, 0.0f, 0.0f, BIGF};        // forces d^2 = BIG
    }
    *(v4f*)(embA + (size_t)idx * 4) = ea;
    *(v4f*)(embB + (size_t)idx * 4) = eb;
}

// One wave32 per (sample, direction, 16-row tile).
// Inner loop: load_b64 (B tile), V_WMMA_F32_16X16X4_F32, elementwise min.
__global__ void sdfch_chamfer(const int* __restrict__ counts,
                              const float* __restrict__ embA,
                              const float* __restrict__ embB,
                              float* __restrict__ accum) {
    int bid  = blockIdx.x;
    int tile = bid & (PMAX / 16 - 1);
    int dir  = (bid >> 9) & 1;
    int b    = bid >> 10;

    int slot1 = b * 2 + dir;
    int slot2 = b * 2 + (1 - dir);
    int c1 = min(counts[slot1], PMAX);
    int c2 = min(counts[slot2], PMAX);
    if (tile * 16 >= c1 || c2 <= 0) return;        // block-uniform exit

    int lane = threadIdx.x;
    int m    = lane & 15;
    int half = lane >> 4;                          // K-pair: 0 -> K{0,1}, 1 -> K{2,3}

    int M = tile * 16 + m;
    v2f a = *(const v2f*)(embA + (((size_t)slot1 * PMAX + M) * 4) + half * 2);
    const float* bbase = embB + (size_t)slot2 * PMAX * 4 + (size_t)(half * 2);

    v8f runmin;
#pragma unroll
    for (int r = 0; r < 8; ++r) runmin[r] = 3.0e37f;

    int nT = (c2 + 15) >> 4;
#pragma unroll 4
    for (int t = 0; t < nT; ++t) {
        v2f bb = *(const v2f*)(bbase + (size_t)(t * 16 + m) * 4);
        v8f cz = {};
        v8f d = __builtin_amdgcn_wmma_f32_16x16x4_f32(
            false, a, false, bb, (short)0, cz, false, false);
#pragma unroll
        for (int r = 0; r < 8; ++r) runmin[r] = fminf(runmin[r], d[r]);
    }

    // min over the 16 columns held across each 16-lane half
#pragma unroll
    for (int off = 1; off < 16; off <<= 1) {
#pragma unroll
        for (int r = 0; r < 8; ++r)
            runmin[r] = fminf(runmin[r], __shfl_xor(runmin[r], off, 32));
    }

    // lane 0 holds rows 0..7 mins, lane 16 holds rows 8..15 mins
    if (m == 0) {
        float s = 0.0f;
#pragma unroll
        for (int r = 0; r < 8; ++r) {
            int row = tile * 16 + half * 8 + r;
            if (row < c1) s += sqrtf(runmin[r]);
        }
        atomicAdd(&accum[dir * 4 + b], s);
    }
}

__global__ void sdfch_l1(const float* __restrict__ pred,
                         const float* __restrict__ gt,
                         float* __restrict__ accum) {
    __shared__ float red[256];
    int idx = blockIdx.x * 256 + threadIdx.x;
    int b   = idx / (HH * WW);
    red[threadIdx.x] = fabsf(pred[idx] - gt[idx]);
    __syncthreads();
    for (int s = 128; s > 0; s >>= 1) {
        if (threadIdx.x < s) red[threadIdx.x] += red[threadIdx.x + s];
        __syncthreads();
    }
    if (threadIdx.x == 0) atomicAdd(&accum[8 + b], red[0]);
}

__global__ void sdfch_finalize(const int* __restrict__ counts,
                               const float* __restrict__ accum,
                               float* __restrict__ out) {
    float total = 0.0f;
    for (int b = 0; b < BB; ++b) {
        int n1 = min(counts[b * 2], PMAX);
        int n2 = min(counts[b * 2 + 1], PMAX);
        float mean1 = accum[b]     / fmaxf((float)n1, 1.0f);
        float mean2 = accum[4 + b] / fmaxf((float)n2, 1.0f);
        float ch  = (n1 > 0 && n2 > 0) ? fabsf(-mean1 + mean2) : 0.0f;
        float sdf = accum[8 + b] / (float)(HH * WW);
        total += ch + sdf;
    }
    out[0] = total / (float)BB;
}

extern "C" void kernel_launch(void* const* d_in, const int* in_sizes, int n_in,
                              void* d_out, int out_size, void* d_ws, size_t ws_size,
                              hipStream_t stream) {
    const float* pred = (const float*)d_in[0];
    const float* gt   = (const float*)d_in[1];
    int*   counts = (int*)d_ws;
    float* accum  = (float*)d_ws + 8;
    float* pts    = (float*)d_ws + 32;
    float* embA   = (float*)d_ws + 32 + 8 * PMAX * 2;
    float* embB   = embA + 8 * PMAX * 4;
    float* out    = (float*)d_out;

    sdfch_init<<<1, 32, 0, stream>>>(counts, accum);

    const int CAND_TOTAL = 8 * ((HH - 1) * WW + HH * (WW - 1));  // 1,044,480
    sdfch_extract<<<(CAND_TOTAL + 255) / 256, 256, 0, stream>>>(pred, gt, counts, pts);

    sdfch_embed<<<8 * PMAX / 256, 256, 0, stream>>>(counts, pts, embA, embB);

    sdfch_chamfer<<<BB * 2 * (PMAX / 16), 32, 0, stream>>>(counts, embA, embB, accum);

    sdfch_l1<<<BB * HH * WW / 256, 256, 0, stream>>>(pred, gt, accum);

    sdfch_finalize<<<1, 1, 0, stream>>>(counts, accum, out);
}